// FGVCResnet_73469710565717
// MI455X (gfx1250) — compile-verified
//
#include <hip/hip_runtime.h>
#include <cstdint>

typedef __attribute__((ext_vector_type(2))) float v2f;
typedef __attribute__((ext_vector_type(4))) float v4f;
typedef __attribute__((ext_vector_type(8))) float v8f;

#define CCH   2048   // channels (reduction dim)
#define HW    1600   // 40*40 spatial
#define H_    40
#define NCLS  200
#define NTT   13     // ceil(200/16) class tiles, all held in registers per wave
#define NPAD  208    // 13*16 padded class count
#define KC    128    // K-chunk staged in LDS: 128*208*4B = 104 KB (of 320 KB/WGP)
#define KS    20
#define ALPHA 0.1f
#define BATCH 32

// ---------------------------------------------------------------------------
// Kernel 1: cam[b][k][hw] = sum_c F[b][c][hw] * W[c][k]
// N-blocked register tiling: each wave owns one 16-row hw tile and ALL 13
// class tiles (13 x v8f = 104 acc VGPRs), so every A fragment loaded from
// global is reused 13x from registers -> A streamed from HBM exactly once.
// W chunk in LDS with K interleaved: [KC/4][NPAD][4] so each B fragment is
// one aligned ds_load_b64 straight into an even VGPR pair (WMMA source).
// ---------------------------------------------------------------------------
__global__ __launch_bounds__(256) void cam_gemm_kernel(const float* __restrict__ F,
                                                       const float* __restrict__ W,
                                                       float* __restrict__ cam) {
  extern __shared__ float sW[];                 // [KC/4][NPAD][4] = 104 KB
  const int b  = blockIdx.x;
  const int mt = blockIdx.y * 8 + (threadIdx.x >> 5);  // hw tile index (0..103)
  const bool active = (mt < 100);
  const float* Fb = F + (size_t)b * CCH * HW;

  const int lane = threadIdx.x & 31;
  const int hl   = lane >> 4;     // 0: lanes 0-15 (K0/K1), 1: lanes 16-31 (K2/K3)
  const int l15  = lane & 15;
  const int hwbase = mt * 16;

  v8f acc[NTT];
  #pragma unroll
  for (int nt = 0; nt < NTT; ++nt) acc[nt] = v8f{};

  for (int chunk = 0; chunk < CCH / KC; ++chunk) {
    const int c0 = chunk * KC;
    // cooperative stage: thread packs 4 consecutive K-rows of one class column
    // into a contiguous 16B LDS slot -> ds_store_b128, coalesced global reads
    for (int i = threadIdx.x; i < (KC / 4) * NPAD; i += 256) {
      const int rg = i / NPAD;          // K-group (4 rows)
      const int k  = i - rg * NPAD;     // class column
      v4f v = {};
      if (k < NCLS) {
        const float* wp = W + (size_t)(c0 + rg * 4) * NCLS + k;
        #pragma unroll
        for (int j = 0; j < 4; ++j) v[j] = wp[(size_t)j * NCLS];
      }
      *(v4f*)(sW + (size_t)i * 4) = v;  // sW[rg][k][0..3]
    }
    __syncthreads();

    if (active) {
      // A frag: lane l15 -> M=l15 (hw), VGPR0=K(2*hl), VGPR1=K(2*hl+1)
      const float* ap = Fb + (size_t)(c0 + 2 * hl) * HW + hwbase + l15;
      // B frag base: class col l15, K-pair selected by hl
      const float* bp = sW + (size_t)l15 * 4 + hl * 2;
      for (int cs = 0; cs < KC; cs += 4) {
        v2f a;
        a.x = ap[0];  a.y = ap[HW];
        ap += 4 * HW;
        #pragma unroll
        for (int nt = 0; nt < NTT; ++nt) {
          // one ds_load_b64: {K(2hl), K(2hl+1)} for class nt*16+l15
          v2f bm = *(const v2f*)(bp + nt * 64);
          acc[nt] = __builtin_amdgcn_wmma_f32_16x16x4_f32(false, a, false, bm,
                                                          (short)0, acc[nt],
                                                          false, false);
        }
        bp += NPAD * 4;
      }
    }
    __syncthreads();
  }

  if (active) {
    // C/D layout: VGPR j, lanes 0-15 -> (M=j, N=l15); lanes 16-31 -> (M=8+j)
    #pragma unroll
    for (int nt = 0; nt < NTT; ++nt) {
      const int k = nt * 16 + l15;
      if (k < NCLS) {
        float* cp = cam + ((size_t)b * NCLS + k) * HW + hwbase + 8 * hl;
        #pragma unroll
        for (int j = 0; j < 8; ++j) cp[j] = acc[nt][j];   // -> 2x global_store_b128
      }
    }
  }
}

// ---------------------------------------------------------------------------
// JAX threefry2x32 (20 rounds, 5 key injections) - bit-exact with jax.random
// ---------------------------------------------------------------------------
__device__ __forceinline__ uint32_t rotl32(uint32_t x, int r) {
  return (x << r) | (x >> (32 - r));
}

__device__ __forceinline__ void threefry2x32(uint32_t k0, uint32_t k1,
                                             uint32_t c0, uint32_t c1,
                                             uint32_t* o0, uint32_t* o1) {
  const uint32_t k2 = k0 ^ k1 ^ 0x1BD11BDAu;
  uint32_t x0 = c0 + k0, x1 = c1 + k1;
  const int rotA[4] = {13, 15, 26, 6};
  const int rotB[4] = {17, 29, 16, 24};
#define TF_R4(rot) { _Pragma("unroll") \
  for (int i = 0; i < 4; ++i) { x0 += x1; x1 = rotl32(x1, rot[i]); x1 ^= x0; } }
  TF_R4(rotA); x0 += k1; x1 += k2 + 1u;
  TF_R4(rotB); x0 += k2; x1 += k0 + 2u;
  TF_R4(rotA); x0 += k0; x1 += k1 + 3u;
  TF_R4(rotB); x0 += k1; x1 += k2 + 4u;
  TF_R4(rotA); x0 += k2; x1 += k0 + 5u;
#undef TF_R4
  *o0 = x0; *o1 = x1;
}

__device__ __forceinline__ float u01(uint32_t bits) {
  return __uint_as_float((bits >> 9) | 0x3F800000u) - 1.0f;
}
__device__ __forceinline__ float clamp01(float v) { return fminf(fmaxf(v, 0.0f), 1.0f); }

// ---------------------------------------------------------------------------
// Kernel 2 (fused stats + diversify + mean): one block per (b,k). The cam row
// (1600 f32 = 6.4 KB) is staged in LDS once, so cam is read from global
// exactly once: maxes, threefry Bernoulli flags, masked mean all from LDS.
// ---------------------------------------------------------------------------
__global__ __launch_bounds__(256) void cam_diversify_mean_kernel(const float* __restrict__ cam,
                                                                 float* __restrict__ out) {
  __shared__ float srow[HW];
  __shared__ float sred[4][256];
  __shared__ float ssum[256];

  const int bk = blockIdx.x;
  const float* row = cam + (size_t)bk * HW;

  float pm[4] = {-3.4e38f, -3.4e38f, -3.4e38f, -3.4e38f};
  for (int hw = threadIdx.x; hw < HW; hw += 256) {
    const float x = row[hw];
    srow[hw] = x;
    const int h = hw / H_, w = hw % H_;
    const int p = (h / KS) * 2 + (w / KS);
    pm[p] = fmaxf(pm[p], x);
  }
  #pragma unroll
  for (int p = 0; p < 4; ++p) sred[p][threadIdx.x] = pm[p];
  __syncthreads();
  for (int off = 128; off > 0; off >>= 1) {
    if (threadIdx.x < off)
      #pragma unroll
      for (int p = 0; p < 4; ++p)
        sred[p][threadIdx.x] = fmaxf(sred[p][threadIdx.x], sred[p][threadIdx.x + off]);
    __syncthreads();
  }
  float pmx[4];
  #pragma unroll
  for (int p = 0; p < 4; ++p) pmx[p] = sred[p][0];
  const float am = fmaxf(fmaxf(pmx[0], pmx[1]), fmaxf(pmx[2], pmx[3]));

  // kp, kq = jax.random.split(jax.random.key(42)); key(42) -> (0, 42)
  // split: counts iota(4) -> blocks (0,2),(1,3); kp=(o0_a,o0_b), kq=(o1_a,o1_b)
  uint32_t a0, a1, b0, b1, o0, o1;
  threefry2x32(0u, 42u, 0u, 2u, &a0, &a1);
  threefry2x32(0u, 42u, 1u, 3u, &b0, &b1);

  // uniform(kp, 6400): counts split in half; elem j<3200 -> word0 of block j
  const uint32_t ip = (bk < 3200) ? (uint32_t)bk : (uint32_t)(bk - 3200);
  threefry2x32(a0, b0, ip, ip + 3200u, &o0, &o1);
  const int p_peak = u01((bk < 3200) ? o0 : o1) < clamp01(am);

  int p_patch[4];
  #pragma unroll
  for (int p = 0; p < 4; ++p) {
    const uint32_t j  = (uint32_t)bk * 4u + (uint32_t)p;   // (b,K,l,m) row-major
    const uint32_t iq = (j < 12800u) ? j : j - 12800u;
    threefry2x32(a1, b1, iq, iq + 12800u, &o0, &o1);
    p_patch[p] = u01((j < 12800u) ? o0 : o1) < clamp01(pmx[p]);
  }

  float sum = 0.0f;
  for (int hw = threadIdx.x; hw < HW; hw += 256) {
    const float x = srow[hw];
    const int h = hw / H_, w = hw % H_;
    const int p = (h / KS) * 2 + (w / KS);
    // at peak positions bc = p_peak; elsewhere bc = p_patch of the patch
    const int sup = (x == am) ? p_peak : p_patch[p];
    sum += sup ? x * ALPHA : x;
  }
  ssum[threadIdx.x] = sum;
  __syncthreads();
  for (int off = 128; off > 0; off >>= 1) {
    if (threadIdx.x < off) ssum[threadIdx.x] += ssum[threadIdx.x + off];
    __syncthreads();
  }
  if (threadIdx.x == 0) out[bk] = ssum[0] * (1.0f / HW);
}

// ---------------------------------------------------------------------------
extern "C" void kernel_launch(void* const* d_in, const int* in_sizes, int n_in,
                              void* d_out, int out_size, void* d_ws, size_t ws_size,
                              hipStream_t stream) {
  const float* F = (const float*)d_in[0];   // [32, 2048, 40, 40]
  const float* W = (const float*)d_in[1];   // [2048, 200]
  float* out = (float*)d_out;               // [32, 200]
  float* cam = (float*)d_ws;                // 32*200*1600 f32 (~41 MB scratch)

  // 32 batches x 13 hw-tile groups (8 waves/WG, waves 100..103 idle)
  dim3 g1(BATCH, 13);
  cam_gemm_kernel<<<g1, 256, KC * NPAD * sizeof(float), stream>>>(F, W, cam);
  cam_diversify_mean_kernel<<<BATCH * NCLS, 256, 0, stream>>>(cam, out);
}